// SSIMLoss_43353399886397
// MI455X (gfx1250) — compile-verified
//
#include <hip/hip_runtime.h>

// ---------------------------------------------------------------------------
// SSIM (uniform 11x11 window) on MI455X / gfx1250.
// - Box filter as banded-matrix GEMM on the matrix pipe: V_WMMA_F32_16X16X4_F32
//   (fp32 in / fp32 accumulate preserves the uxx - ux^2 cancellation).
// - Input tiles staged global->LDS by the Tensor Data Mover (tensor_load_to_lds,
//   TENSORcnt), one 28x32 fp32 tile per input per wave; TDM OOB reads return 0,
//   giving exact zero halo padding at image edges.
// - One wave32 per 16x16 output tile; per-wave private LDS; fp64 atomic reduce.
// Roofline: 64 MB input @ 23.3 TB/s ~ 2.8 us floor; whole working set fits in
// the 192 MB L2 so the 2.6x tile-halo amplification stays on-chip.
// ---------------------------------------------------------------------------

typedef __attribute__((ext_vector_type(2))) float        v2f;
typedef __attribute__((ext_vector_type(8))) float        v8f;
typedef __attribute__((ext_vector_type(4))) unsigned int u32x4;
typedef __attribute__((ext_vector_type(8))) int          i32x8;
typedef __attribute__((ext_vector_type(4))) int          i32x4;

#define HW      512
#define WIN     11
#define VALID   (HW - WIN + 1)          // 502
#define TILE    16
#define NTILES  32                      // ceil(502/16)
#define NIMG    32
#define KPAD    28                      // 26 rounded up to multiple of 4
#define LDSW    32                      // padded LDS row width (floats)
// per-wave LDS floats: P[28][32] + T[28][32] + TQ[5][16][32]
#define WAVE_LDS (KPAD * LDSW * 2 + 5 * 16 * LDSW)   // 4352 floats
#define WAVES_PER_BLOCK 4
#define TOTAL_TILES (NIMG * NTILES * NTILES)         // 32768

#if defined(__has_builtin)
#if __has_builtin(__builtin_amdgcn_tensor_load_to_lds)
#define USE_TDM 1
#endif
#endif

// 16x26 band matrix: row r carries 1/11 at columns r..r+10 (zero for padded K).
__device__ __forceinline__ float wband(int r, int k) {
  return ((unsigned)(k - r) <= (unsigned)(WIN - 1)) ? (1.0f / (float)WIN) : 0.0f;
}

#ifdef USE_TDM
// Issue one TDM DMA: 28x32-element fp32 tile starting at (oy,ox) of a 512x512
// image -> contiguous LDS rows of 32 floats at lds_off. tensor_dim is the
// REMAINING extent from the tile origin so reads past the image edge return 0.
__device__ __forceinline__ void tdm_load_2d(unsigned lds_off, const float* gbase,
                                            int oy, int ox) {
  unsigned long long ga =
      (unsigned long long)(uintptr_t)(gbase + (size_t)oy * HW + ox);
  const int dim0 = HW - ox;    // x extent left of tensor
  const int dim1 = HW - oy;    // y extent left of tensor
  u32x4 g0;
  g0[0] = 1u;                                              // count=1, user D#
  g0[1] = lds_off;                                         // LDS byte address
  g0[2] = (unsigned)(ga & 0xFFFFFFFFull);                  // global_addr lo
  g0[3] = (unsigned)((ga >> 32) & 0x1FFFFFFull) | (2u << 30); // addr hi | type=2
  i32x8 g1;
  g1[0] = (2 << 16);                                       // data_size=4B
  g1[1] = (dim0 & 0xFFFF) << 16;                           // tensor_dim0 lo16
  g1[2] = ((dim0 >> 16) & 0xFFFF) | ((dim1 & 0xFFFF) << 16);
  g1[3] = ((dim1 >> 16) & 0xFFFF) | (LDSW << 16);          // tile_dim0 = 32
  g1[4] = KPAD;                                            // tile_dim1=28, dim2=0
  g1[5] = HW;                                              // dim0_stride = 512
  g1[6] = 0;
  g1[7] = 0;                                               // dim1_stride unused (2D)
  i32x4 z4 = {0, 0, 0, 0};
#if __clang_major__ >= 23
  i32x8 z8 = {0, 0, 0, 0, 0, 0, 0, 0};
  __builtin_amdgcn_tensor_load_to_lds(g0, g1, z4, z4, z8, 0);
#else
  __builtin_amdgcn_tensor_load_to_lds(g0, g1, z4, z4, 0);
#endif
}
#endif

__global__ void __launch_bounds__(WAVES_PER_BLOCK * 32)
ssim_wmma_kernel(const float* __restrict__ pred,
                 const float* __restrict__ targ,
                 double* __restrict__ accum) {
  extern __shared__ float smem[];
  const int lane = threadIdx.x & 31;
  const int wave = threadIdx.x >> 5;
  float* Pl = smem + wave * WAVE_LDS;   // pred patch  [28][32]
  float* Tl = Pl + KPAD * LDSW;         // target patch[28][32]
  float* TQ = Tl + KPAD * LDSW;         // stage-1 out [5][16][32]

  const int tile = blockIdx.x * WAVES_PER_BLOCK + wave;   // exact cover
  const int b    = tile >> 10;                            // / (32*32)
  const int rem  = tile & 1023;
  const int oy   = (rem >> 5) * TILE;                     // valid-output origin
  const int ox   = (rem & 31) * TILE;

  const float* pimg = pred + (size_t)b * HW * HW;
  const float* timg = targ + (size_t)b * HW * HW;

  // ---- stage 0: stage the 28x32 halo patches into LDS
#ifdef USE_TDM
  // Generic->LDS aperture: low 32 pointer bits are the DS-relative byte addr.
  const unsigned p_off = (unsigned)(uintptr_t)Pl;
  const unsigned t_off = (unsigned)(uintptr_t)Tl;
  tdm_load_2d(p_off, pimg, oy, ox);
  tdm_load_2d(t_off, timg, oy, ox);
  __builtin_amdgcn_s_wait_tensorcnt((short)0);
#else
  for (int idx = lane; idx < KPAD * LDSW; idx += 32) {
    const int r = idx >> 5, c = idx & (LDSW - 1);
    const int y = oy + r, x = ox + c;
    const bool ok = (y < HW) && (x < HW);
    const size_t g = (size_t)min(y, HW - 1) * HW + min(x, HW - 1);
    Pl[idx] = ok ? pimg[g] : 0.0f;
    Tl[idx] = ok ? timg[g] : 0.0f;
  }
#endif
  __syncthreads();

  const int n     = lane & 15;             // A row / B column / C,D column
  const int koff  = (lane < 16) ? 0 : 2;   // K sub-index split across lane halves
  const int mbase = (lane >> 4) << 3;      // C/D row base (M = i + mbase)

  // Constant band-filter A operand, shared by both passes (pure VALU, no mem).
  v2f aw[7];
#pragma unroll
  for (int kc = 0; kc < 7; ++kc) {
    const int k0 = kc * 4 + koff;
    aw[kc].x = wband(n, k0);
    aw[kc].y = wband(n, k0 + 1);
  }

  // ---- stage 1: vertical filter  T_q(16x26) = Wv(16x26) x Q(26x26)
  // N split into two 16-wide tiles; K=26 padded to 28 -> 7 chunks of 4.
  // Padded K rows 26..27 carry finite data but wband == 0 there.
  for (int ntile = 0; ntile < 2; ++ntile) {
    const int cb = ntile * 16;
    v8f acc0 = {}, acc1 = {}, acc2 = {}, acc3 = {}, acc4 = {};
#pragma unroll
    for (int kc = 0; kc < 7; ++kc) {
      const int k0 = kc * 4 + koff;
      const float p0 = Pl[k0 * LDSW + cb + n];
      const float p1 = Pl[(k0 + 1) * LDSW + cb + n];
      const float t0 = Tl[k0 * LDSW + cb + n];
      const float t1 = Tl[(k0 + 1) * LDSW + cb + n];
      v2f bb;
      bb.x = p0;      bb.y = p1;       // ux
      acc0 = __builtin_amdgcn_wmma_f32_16x16x4_f32(false, aw[kc], false, bb, (short)0, acc0, false, false);
      bb.x = t0;      bb.y = t1;       // uy
      acc1 = __builtin_amdgcn_wmma_f32_16x16x4_f32(false, aw[kc], false, bb, (short)0, acc1, false, false);
      bb.x = p0 * p0; bb.y = p1 * p1;  // uxx
      acc2 = __builtin_amdgcn_wmma_f32_16x16x4_f32(false, aw[kc], false, bb, (short)0, acc2, false, false);
      bb.x = t0 * t0; bb.y = t1 * t1;  // uyy
      acc3 = __builtin_amdgcn_wmma_f32_16x16x4_f32(false, aw[kc], false, bb, (short)0, acc3, false, false);
      bb.x = p0 * t0; bb.y = p1 * t1;  // uxy
      acc4 = __builtin_amdgcn_wmma_f32_16x16x4_f32(false, aw[kc], false, bb, (short)0, acc4, false, false);
    }
#pragma unroll
    for (int i = 0; i < 8; ++i) {
      const int row = (mbase + i) * LDSW + cb + n;   // C/D layout -> row major
      TQ[0 * 512 + row] = acc0[i];
      TQ[1 * 512 + row] = acc1[i];
      TQ[2 * 512 + row] = acc2[i];
      TQ[3 * 512 + row] = acc3[i];
      TQ[4 * 512 + row] = acc4[i];
    }
  }
  __syncthreads();

  // ---- stage 2: horizontal filter  Out^T(16x16) = Wv(16x26) x T_q^T(26x16)
  // B operand of T^T is T[n][k], T[n][k+1]: contiguous LDS reads; padded K
  // columns 26..27 are consumed only where wband == 0.
  v8f outq[5];
#pragma unroll
  for (int q = 0; q < 5; ++q) {
    v8f c = {};
#pragma unroll
    for (int kc = 0; kc < 7; ++kc) {
      const int k0 = kc * 4 + koff;
      v2f bb;
      bb.x = TQ[q * 512 + n * LDSW + k0];
      bb.y = TQ[q * 512 + n * LDSW + k0 + 1];
      c = __builtin_amdgcn_wmma_f32_16x16x4_f32(false, aw[kc], false, bb, (short)0, c, false, false);
    }
    outq[q] = c;
  }

  // ---- epilogue: per-pixel SSIM, valid-region mask, wave reduce, f64 atomic
  const float C1c = 1e-4f;                 // (0.01*1.0)^2
  const float C2c = 9e-4f;                 // (0.03*1.0)^2
  const float cov = 121.0f / 120.0f;       // NP/(NP-1)
  float s = 0.0f;
#pragma unroll
  for (int i = 0; i < 8; ++i) {
    const int m = mbase + i;               // Out^T[m][n] == pixel(oy+n, ox+m)
    const int y = oy + n, x = ox + m;
    const float ux = outq[0][i], uy = outq[1][i];
    const float vx  = cov * (outq[2][i] - ux * ux);
    const float vy  = cov * (outq[3][i] - uy * uy);
    const float vxy = cov * (outq[4][i] - ux * uy);
    const float A1 = 2.0f * ux * uy + C1c;
    const float A2 = 2.0f * vxy + C2c;
    const float B1 = ux * ux + uy * uy + C1c;
    const float B2 = vx + vy + C2c;
    const float S = (A1 * A2) / (B1 * B2);
    if (y < VALID && x < VALID) s += S;
  }
#pragma unroll
  for (int off = 16; off > 0; off >>= 1)
    s += __shfl_xor(s, off, 32);
  if (lane == 0)
    atomicAdd(accum, (double)s);
}

__global__ void ssim_zero_kernel(double* acc) {
  if (threadIdx.x == 0) acc[0] = 0.0;
}

__global__ void ssim_fin_kernel(const double* acc, float* out) {
  if (threadIdx.x == 0) {
    const double denom = (double)NIMG * (double)VALID * (double)VALID;
    out[0] = (float)(1.0 - acc[0] / denom);
  }
}

extern "C" void kernel_launch(void* const* d_in, const int* in_sizes, int n_in,
                              void* d_out, int out_size, void* d_ws, size_t ws_size,
                              hipStream_t stream) {
  const float* pred = (const float*)d_in[0];
  const float* targ = (const float*)d_in[1];
  double* acc = (double*)d_ws;
  float* out = (float*)d_out;

  const size_t shmem = (size_t)WAVES_PER_BLOCK * WAVE_LDS * sizeof(float); // 69,632 B
  (void)hipFuncSetAttribute((const void*)ssim_wmma_kernel,
                            hipFuncAttributeMaxDynamicSharedMemorySize,
                            (int)shmem);

  ssim_zero_kernel<<<1, 64, 0, stream>>>(acc);
  ssim_wmma_kernel<<<TOTAL_TILES / WAVES_PER_BLOCK, WAVES_PER_BLOCK * 32,
                     shmem, stream>>>(pred, targ, acc);
  ssim_fin_kernel<<<1, 64, 0, stream>>>(acc, out);
}